// imageCaptionModel_17892833755423
// MI455X (gfx1250) — compile-verified
//
#include <hip/hip_runtime.h>
#include <hip/hip_bf16.h>
#include <math.h>
#include <stdint.h>

// Problem dims (fixed by the reference)
#define VV    10000
#define EE    512
#define NCNN_ 2048
#define HH    512
#define LL    2
#define BB    128
#define TT    25

typedef float v2f __attribute__((ext_vector_type(2)));
typedef float v8f __attribute__((ext_vector_type(8)));

// D = A(16x4,f32) * B(4x16,f32) + C(16x16,f32)  -- exact fp32 matrix core op
__device__ __forceinline__ v8f wmma4(v2f a, v2f b, v8f c) {
  return __builtin_amdgcn_wmma_f32_16x16x4_f32(false, a, false, b, (short)0, c,
                                               false, false);
}

// ---------------------------------------------------------------------------
// Register-blocked wave-level GEMM accumulators (fp32, K multiple of 4).
// Lane layout per CDNA5 ISA 7.12.2 (wave32):
//   A: lane l -> row m = l&15, K = 2*(l>>4)+v          (v = vgpr 0/1)
//   B: lane l -> col n = l&15, K = 2*(l>>4)+v
//   C: lane l, vgpr v -> row m = v + 8*(l>>4), col n = l&15
// ---------------------------------------------------------------------------

// B stored "normal": B[k][n] at Bm + k*ldb + n   (GRU weights [K, H])
template <int R, int C>
__device__ __forceinline__ void mm_nn_block(v8f (&acc)[R][C],
    const float* __restrict__ A, int lda,
    const float* __restrict__ Bm, int ldb, int K, int lane) {
  const int m  = lane & 15;
  const int kh = (lane >> 4) << 1;               // 0 or 2
  const float* ap[R];
  const float* bp[C];
#pragma unroll
  for (int r = 0; r < R; ++r) ap[r] = A + (r * 16 + m) * lda + kh;
#pragma unroll
  for (int c = 0; c < C; ++c) bp[c] = Bm + kh * ldb + c * 16 + m;
#pragma unroll 2
  for (int k = 0; k < K; k += 4) {
    v2f af[R], bf[C];
#pragma unroll
    for (int r = 0; r < R; ++r) af[r] = *(const v2f*)(ap[r] + k);
#pragma unroll
    for (int c = 0; c < C; ++c) {
      bf[c].x = bp[c][k * ldb];
      bf[c].y = bp[c][(k + 1) * ldb];
    }
#pragma unroll
    for (int r = 0; r < R; ++r)
#pragma unroll
      for (int c = 0; c < C; ++c) acc[r][c] = wmma4(af[r], bf[c], acc[r][c]);
  }
}

// B from a transposed weight: B[k][n] = Wt[n][k] at Wt + n*ldt + k
template <int R, int C>
__device__ __forceinline__ void mm_nt_block(v8f (&acc)[R][C],
    const float* __restrict__ A, int lda,
    const float* __restrict__ Wt, int ldt, int K, int lane) {
  const int m  = lane & 15;
  const int kh = (lane >> 4) << 1;
  const float* ap[R];
  const float* bp[C];
#pragma unroll
  for (int r = 0; r < R; ++r) ap[r] = A + (r * 16 + m) * lda + kh;
#pragma unroll
  for (int c = 0; c < C; ++c) bp[c] = Wt + (c * 16 + m) * ldt + kh;
#pragma unroll 2
  for (int k = 0; k < K; k += 4) {
    v2f af[R], bf[C];
#pragma unroll
    for (int r = 0; r < R; ++r) af[r] = *(const v2f*)(ap[r] + k);
#pragma unroll
    for (int c = 0; c < C; ++c) bf[c] = *(const v2f*)(bp[c] + k);
#pragma unroll
    for (int r = 0; r < R; ++r)
#pragma unroll
      for (int c = 0; c < C; ++c) acc[r][c] = wmma4(af[r], bf[c], acc[r][c]);
  }
}

// A formed on the fly as r*s (GRU candidate), B normal layout.
template <int R, int C>
__device__ __forceinline__ void mm_rs_block(v8f (&acc)[R][C],
    const float* __restrict__ Rr, const float* __restrict__ S, int lda,
    const float* __restrict__ Bm, int ldb, int K, int lane) {
  const int m  = lane & 15;
  const int kh = (lane >> 4) << 1;
  const float* rp[R];
  const float* sp[R];
  const float* bp[C];
#pragma unroll
  for (int r = 0; r < R; ++r) {
    rp[r] = Rr + (r * 16 + m) * lda + kh;
    sp[r] = S + (r * 16 + m) * lda + kh;
  }
#pragma unroll
  for (int c = 0; c < C; ++c) bp[c] = Bm + kh * ldb + c * 16 + m;
#pragma unroll 2
  for (int k = 0; k < K; k += 4) {
    v2f af[R], bf[C];
#pragma unroll
    for (int r = 0; r < R; ++r) {
      v2f ra = *(const v2f*)(rp[r] + k);
      v2f sa = *(const v2f*)(sp[r] + k);
      af[r] = ra * sa;
    }
#pragma unroll
    for (int c = 0; c < C; ++c) {
      bf[c].x = bp[c][k * ldb];
      bf[c].y = bp[c][(k + 1) * ldb];
    }
#pragma unroll
    for (int r = 0; r < R; ++r)
#pragma unroll
      for (int c = 0; c < C; ++c) acc[r][c] = wmma4(af[r], bf[c], acc[r][c]);
  }
}

#define ZACC2x2 {{{0.f,0.f,0.f,0.f,0.f,0.f,0.f,0.f},{0.f,0.f,0.f,0.f,0.f,0.f,0.f,0.f}}, \
                 {{0.f,0.f,0.f,0.f,0.f,0.f,0.f,0.f},{0.f,0.f,0.f,0.f,0.f,0.f,0.f,0.f}}}

// ---------------------------------------------------------------------------
// Embedding gather: xs[t][b][:] = emb_table[xTokens[b][t]][:]   (time-major)
// ---------------------------------------------------------------------------
__global__ __launch_bounds__(128) void embed_kernel(
    const int* __restrict__ xTokens, const float* __restrict__ emb,
    float* __restrict__ xs) {
  const int b   = blockIdx.x / TT;
  const int t   = blockIdx.x - b * TT;
  const int tok = xTokens[blockIdx.x];           // xTokens[b*T + t]
  const float4* src = (const float4*)(emb + (size_t)tok * EE);
  float4* dst = (float4*)(xs + ((size_t)t * BB + b) * EE);
  dst[threadIdx.x] = src[threadIdx.x];           // 128 threads x float4 = 512 f
}

// ---------------------------------------------------------------------------
// h0 = tanh(cnn[128,2048] @ win_w[512,2048]^T + win_b) -> s0 and s1 init
// ---------------------------------------------------------------------------
__global__ __launch_bounds__(256) void h0_kernel(
    const float* __restrict__ cnn, const float* __restrict__ win_w,
    const float* __restrict__ win_b, float* __restrict__ s0,
    float* __restrict__ s1) {
  const int lane = threadIdx.x & 31;
  const int wave = blockIdx.x * 8 + (threadIdx.x >> 5);  // 0..63
  const int bt = wave >> 4, ht = wave & 15;
  v8f acc[2][2] = ZACC2x2;
  mm_nt_block<2, 2>(acc, cnn + bt * 32 * NCNN_, NCNN_,
                    win_w + (size_t)(ht * 32) * NCNN_, NCNN_, NCNN_, lane);
  const int n = lane & 15, kh = lane >> 4;
#pragma unroll
  for (int r = 0; r < 2; ++r)
#pragma unroll
    for (int c = 0; c < 2; ++c) {
      const int col = ht * 32 + c * 16 + n;
      const float bias = win_b[col];
#pragma unroll
      for (int v = 0; v < 8; ++v) {
        const int row = bt * 32 + r * 16 + v + 8 * kh;
        const float h = tanhf(acc[r][c][v] + bias);
        s0[row * HH + col] = h;
        s1[row * HH + col] = h;
      }
    }
}

// ---------------------------------------------------------------------------
// GRU gates: u = sig([x,s]@wu + bu), r = sig([x,s]@wr + br), cx = x@w[:E,:]
// ---------------------------------------------------------------------------
__global__ __launch_bounds__(256) void gates_kernel(
    const float* __restrict__ x, const float* __restrict__ s,
    const float* __restrict__ wu, const float* __restrict__ wr,
    const float* __restrict__ w, const float* __restrict__ bu,
    const float* __restrict__ br, float* __restrict__ uo,
    float* __restrict__ ro, float* __restrict__ cxo) {
  const int lane = threadIdx.x & 31;
  const int wave = blockIdx.x * 8 + (threadIdx.x >> 5);
  const int job  = wave >> 6;                    // 0:u  1:r  2:cx
  const int tile = wave & 63;
  const int bt = tile >> 4, ht = tile & 15;      // 4 x 16 macro-tiles of 32x32
  const float* W = (job == 0) ? wu : (job == 1) ? wr : w;
  v8f acc[2][2] = ZACC2x2;
  mm_nn_block<2, 2>(acc, x + bt * 32 * EE, EE, W + ht * 32, HH, EE, lane);
  const int n = lane & 15, kh = lane >> 4;
  if (job < 2) {
    mm_nn_block<2, 2>(acc, s + bt * 32 * HH, HH, W + EE * HH + ht * 32, HH, HH,
                      lane);
    const float* bias = (job == 0) ? bu : br;
    float* out = (job == 0) ? uo : ro;
#pragma unroll
    for (int r = 0; r < 2; ++r)
#pragma unroll
      for (int c = 0; c < 2; ++c) {
        const int col = ht * 32 + c * 16 + n;
        const float bv = bias[col];
#pragma unroll
        for (int v = 0; v < 8; ++v) {
          const float p = acc[r][c][v] + bv;
          out[(bt * 32 + r * 16 + v + 8 * kh) * HH + col] =
              1.0f / (1.0f + __expf(-p));
        }
      }
  } else {
#pragma unroll
    for (int r = 0; r < 2; ++r)
#pragma unroll
      for (int c = 0; c < 2; ++c) {
        const int col = ht * 32 + c * 16 + n;
#pragma unroll
        for (int v = 0; v < 8; ++v)
          cxo[(bt * 32 + r * 16 + v + 8 * kh) * HH + col] = acc[r][c][v];
      }
  }
}

// ---------------------------------------------------------------------------
// Candidate + state update:
//   c = tanh(cx + (r*s_old)@w[E:,:] + b);  s_new = u*s_old + (1-u)*c
// ---------------------------------------------------------------------------
__global__ __launch_bounds__(256) void cand_kernel(
    const float* __restrict__ s_old, const float* __restrict__ uo,
    const float* __restrict__ ro, const float* __restrict__ cxo,
    const float* __restrict__ w, const float* __restrict__ bb,
    float* __restrict__ s_new, float* __restrict__ s_copy) {
  const int lane = threadIdx.x & 31;
  const int wave = blockIdx.x * 8 + (threadIdx.x >> 5);  // 0..63
  const int bt = wave >> 4, ht = wave & 15;
  v8f acc[2][2] = ZACC2x2;
  mm_rs_block<2, 2>(acc, ro + bt * 32 * HH, s_old + bt * 32 * HH, HH,
                    w + EE * HH + ht * 32, HH, HH, lane);
  const int n = lane & 15, kh = lane >> 4;
#pragma unroll
  for (int r = 0; r < 2; ++r)
#pragma unroll
    for (int c = 0; c < 2; ++c) {
      const int col = ht * 32 + c * 16 + n;
      const float bias = bb[col];
#pragma unroll
      for (int v = 0; v < 8; ++v) {
        const int row = bt * 32 + r * 16 + v + 8 * kh;
        const float cc = tanhf(acc[r][c][v] + cxo[row * HH + col] + bias);
        const float uu = uo[row * HH + col];
        const float sv = s_old[row * HH + col];
        const float sn = uu * sv + (1.0f - uu) * cc;
        s_new[row * HH + col] = sn;
        if (s_copy) s_copy[row * HH + col] = sn;  // uniform branch
      }
    }
}

// ---------------------------------------------------------------------------
// Batched projection: logits[b,t,:] = s1_all[t,b,:] @ wout_w^T + wout_b
// Block = 4 waves x 32 rows = 128 rows, sharing one 80-col wout tile staged
// in LDS via CDNA5 async copy (global_load_async_to_lds_b128, ASYNCcnt),
// K-chunked (Kc=64, 20KB/chunk, double-buffered = 40KB LDS) and pipelined:
// wait(chunk k) -> barrier -> issue(chunk k+1) -> WMMA from LDS -> barrier.
// B-operand L2 traffic drops 4x vs per-wave streaming.
// ---------------------------------------------------------------------------
#define KC 64          // K-chunk
#define CSEG 16        // 16-byte segments per 64-float row chunk
__device__ __forceinline__ void issue_async_chunk(
    const float* __restrict__ Wt, int kc, float* dst, int tid) {
  // 80 rows x 64 floats = 1280 x b128 transfers; 128 threads x 10 each.
#pragma unroll
  for (int t = 0; t < 10; ++t) {
    const int e   = tid + t * 128;
    const int row = e >> 4;                      // 0..79
    const int seg = e & 15;                      // 0..15
    const uint64_t g =
        (uint64_t)(uintptr_t)(Wt + (size_t)row * HH + kc * KC + seg * 4);
    const unsigned lo = (unsigned)(uintptr_t)(dst + row * KC + seg * 4);
    asm volatile("global_load_async_to_lds_b128 %0, %1, off"
                 :: "v"(lo), "v"(g) : "memory");
  }
}

__global__ __launch_bounds__(128) void logits_kernel(
    const float* __restrict__ s1_all, const float* __restrict__ wout_w,
    const float* __restrict__ wout_b, float* __restrict__ out) {
  __shared__ __align__(16) float bsm[2][80 * KC];
  const int tid  = threadIdx.x;
  const int lane = tid & 31;
  const int wv   = tid >> 5;                     // wave in block 0..3
  const int rb = blockIdx.x / 125;               // 0..24  (128-row block)
  const int cg = blockIdx.x - rb * 125;          // 0..124 (80-col group)
  const int rg = rb * 4 + wv;                    // this wave's 32-row group
  const float* Ab = s1_all + (size_t)rg * 32 * HH;
  const float* Wt = wout_w + (size_t)cg * 80 * HH;

  const int m   = lane & 15;
  const int khk = (lane >> 4) << 1;
  const float* ap0 = Ab + m * HH + khk;
  const float* ap1 = Ab + (16 + m) * HH + khk;

  issue_async_chunk(Wt, 0, bsm[0], tid);

  v8f acc[2][5] = {};
#pragma unroll
  for (int kc = 0; kc < HH / KC; ++kc) {         // 8 chunks
    asm volatile("s_wait_asynccnt 0x0" ::: "memory");
    __syncthreads();                             // chunk kc visible to all
    if (kc < HH / KC - 1)
      issue_async_chunk(Wt, kc + 1, bsm[(kc + 1) & 1], tid);
    const float* bb = bsm[kc & 1];
    const int ko = kc * KC;
#pragma unroll 2
    for (int k = 0; k < KC; k += 4) {
      v2f af[2], bf[5];
      af[0] = *(const v2f*)(ap0 + ko + k);
      af[1] = *(const v2f*)(ap1 + ko + k);
#pragma unroll
      for (int c = 0; c < 5; ++c)
        bf[c] = *(const v2f*)(bb + (c * 16 + m) * KC + khk + k);
#pragma unroll
      for (int r = 0; r < 2; ++r)
#pragma unroll
        for (int c = 0; c < 5; ++c)
          acc[r][c] = wmma4(af[r], bf[c], acc[r][c]);
    }
    __syncthreads();                             // before buffer reuse
  }

  const int n = lane & 15, kh = lane >> 4;
#pragma unroll
  for (int c = 0; c < 5; ++c) {
    const int col = cg * 80 + c * 16 + n;
    const float bias = wout_b[col];
#pragma unroll
    for (int r = 0; r < 2; ++r)
#pragma unroll
      for (int v = 0; v < 8; ++v) {
        const int row = rg * 32 + r * 16 + v + 8 * kh;  // row = t*B + b
        const int t = row >> 7;                         // / 128
        const int b = row & 127;
        out[((size_t)(b * TT + t)) * VV + col] = acc[r][c][v] + bias;
      }
  }
}

// Final states -> d_out tail: [L=2, B, H]
__global__ __launch_bounds__(256) void finstate_kernel(
    const float* __restrict__ s0, const float* __restrict__ s1,
    float* __restrict__ out) {
  const int i = blockIdx.x * 256 + threadIdx.x;  // 0 .. B*H-1
  out[i]           = s0[i];
  out[BB * HH + i] = s1[i];
}

// ---------------------------------------------------------------------------
extern "C" void kernel_launch(void* const* d_in, const int* in_sizes, int n_in,
                              void* d_out, int out_size, void* d_ws,
                              size_t ws_size, hipStream_t stream) {
  const float* cnn    = (const float*)d_in[0];
  const int*   toks   = (const int*)d_in[1];
  // d_in[2] = is_train (unused by the reference math)
  const float* emb    = (const float*)d_in[3];
  const float* win_w  = (const float*)d_in[4];
  const float* win_b  = (const float*)d_in[5];
  const float* wout_w = (const float*)d_in[6];
  const float* wout_b = (const float*)d_in[7];
  const float* wu[2] = {(const float*)d_in[8],  (const float*)d_in[14]};
  const float* wr[2] = {(const float*)d_in[9],  (const float*)d_in[15]};
  const float* wc[2] = {(const float*)d_in[10], (const float*)d_in[16]};
  const float* bu[2] = {(const float*)d_in[11], (const float*)d_in[17]};
  const float* br[2] = {(const float*)d_in[12], (const float*)d_in[18]};
  const float* bc[2] = {(const float*)d_in[13], (const float*)d_in[19]};

  // Workspace layout (floats): ~14.3 MB total
  float* ws    = (float*)d_ws;
  float* xs    = ws;                              // [T,B,E]
  float* s1all = xs + (size_t)TT * BB * EE;       // [T,B,H]
  float* s0a   = s1all + (size_t)TT * BB * HH;
  float* s0b   = s0a + BB * HH;
  float* s1a   = s0b + BB * HH;
  float* s1b   = s1a + BB * HH;
  float* ubuf  = s1b + BB * HH;
  float* rbuf  = ubuf + BB * HH;
  float* cxbuf = rbuf + BB * HH;

  float* out_logits = (float*)d_out;                           // [B,T,V]
  float* out_state  = out_logits + (size_t)BB * TT * VV;       // [L,B,H]

  embed_kernel<<<BB * TT, 128, 0, stream>>>(toks, emb, xs);
  h0_kernel<<<8, 256, 0, stream>>>(cnn, win_w, win_b, s0a, s1a);

  float* s0c = s0a; float* s0n = s0b;
  float* s1c = s1a; float* s1n = s1b;
  for (int t = 0; t < TT; ++t) {
    const float* xt = xs + (size_t)t * BB * EE;
    // layer 0
    gates_kernel<<<24, 256, 0, stream>>>(xt, s0c, wu[0], wr[0], wc[0], bu[0],
                                         br[0], ubuf, rbuf, cxbuf);
    cand_kernel<<<8, 256, 0, stream>>>(s0c, ubuf, rbuf, cxbuf, wc[0], bc[0],
                                       s0n, nullptr);
    // layer 1 (input = layer-0 new state)
    gates_kernel<<<24, 256, 0, stream>>>(s0n, s1c, wu[1], wr[1], wc[1], bu[1],
                                         br[1], ubuf, rbuf, cxbuf);
    cand_kernel<<<8, 256, 0, stream>>>(s1c, ubuf, rbuf, cxbuf, wc[1], bc[1],
                                       s1n, s1all + (size_t)t * BB * HH);
    float* tmp;
    tmp = s0c; s0c = s0n; s0n = tmp;
    tmp = s1c; s1c = s1n; s1n = tmp;
  }

  logits_kernel<<<3125, 128, 0, stream>>>(s1all, wout_w, wout_b, out_logits);
  finstate_kernel<<<(BB * HH) / 256, 256, 0, stream>>>(s0c, s1c, out_state);
}